// MultiHeadGATSkipLayer_14551349199382
// MI455X (gfx1250) — compile-verified
//
#include <hip/hip_runtime.h>
#include <math.h>

typedef float v2f __attribute__((ext_vector_type(2)));
typedef float v8f __attribute__((ext_vector_type(8)));

#define F_IN   128
#define HC     128
#define NHEAD  8
#define NEG_SLOPE 0.2f
#define LN_EPS 1e-5f

// ---------- helpers ----------
__device__ __forceinline__ unsigned encF(float f) {
    unsigned u = __float_as_uint(f);
    return (u & 0x80000000u) ? ~u : (u | 0x80000000u);   // order-preserving float->uint
}
__device__ __forceinline__ float decF(unsigned k) {
    return __uint_as_float((k & 0x80000000u) ? (k & 0x7FFFFFFFu) : ~k);
}
__device__ __forceinline__ float lrelu(float v) { return v > 0.f ? v : NEG_SLOPE * v; }
__device__ __forceinline__ float eluf(float v)  { return v > 0.f ? v : expm1f(v); }

// ---------- kernel 0: init scratch ----------
__global__ __launch_bounds__(256)
void init_ws(float* __restrict__ gat, float* __restrict__ denom,
             unsigned* __restrict__ emax, int Nn) {
    int i = blockIdx.x * blockDim.x + threadIdx.x;
    if (i < Nn * HC) gat[i] = 0.f;
    if (i < Nn * NHEAD) { denom[i] = 0.f; emax[i] = 0x007FFFFFu; /* enc(-inf) */ }
}

// ---------- kernel 0b: pack [W_gat | W_skip] into (k/2, col, k%2) ----------
// Wp[((k>>1)*256 + col)*2 + (k&1)] = W[k][col]  -> B fragment = one b64 load.
__global__ __launch_bounds__(256)
void pack_w(const float* __restrict__ Wg, const float* __restrict__ Ws,
            float* __restrict__ Wp) {
    int i = blockIdx.x * blockDim.x + threadIdx.x;   // i = k*256 + col, 32768 total
    if (i >= F_IN * 2 * HC) return;
    int k   = i >> 8;
    int col = i & 255;
    float v = (col < HC) ? Wg[k * HC + col] : Ws[k * HC + (col - HC)];
    Wp[((size_t)(k >> 1) * 256 + col) * 2 + (k & 1)] = v;
}

// ---------- kernel 1: fused dual GEMM via f32 WMMA ----------
// One wave computes a 16x64 output slab (4 accumulators); A loaded once per
// k-step and reused across 4 WMMAs. No EXEC masking in the hot loop.
__global__ __launch_bounds__(256)
void gemm_dual(const float* __restrict__ x,
               const float* __restrict__ Wp,
               float* __restrict__ h, float* __restrict__ sk, int Nn)
{
    int wid = blockIdx.x * 8 + (threadIdx.x >> 5);
    int Mtiles = (Nn + 15) >> 4;
    int mt  = wid >> 2;        // row tile
    int nt4 = wid & 3;         // 64-col slab: 0..1 -> gat, 2..3 -> skip
    if (mt >= Mtiles) return;

    int m0 = mt << 4;
    int lane = threadIdx.x & 31;
    int lm = lane & 15;        // m (A) / n (B,C) within tile
    int lh = lane >> 4;        // half-wave selector
    int m  = m0 + lm;
    int mc = m < Nn ? m : (Nn - 1);          // clamp: no branches, unstored rows
    const float* xrow = x + (size_t)mc * F_IN;

    int colbase = nt4 * 64 + lm;             // + j*16 per sub-tile, 0..255

    v8f acc0 = {0.f,0.f,0.f,0.f,0.f,0.f,0.f,0.f};
    v8f acc1 = acc0, acc2 = acc0, acc3 = acc0;

    #pragma unroll 4
    for (int k0 = 0; k0 < F_IN; k0 += 4) {
        int ka = k0 + (lh << 1);             // even k for this half-wave
        v2f a = *(const v2f*)(xrow + ka);    // (k, k+1) pair, b64
        const float* wp = Wp + ((size_t)(ka >> 1) * 256 + colbase) * 2;
        v2f b0 = *(const v2f*)(wp +  0 * 32);   // col + 0
        v2f b1 = *(const v2f*)(wp +  1 * 32);   // col + 16
        v2f b2 = *(const v2f*)(wp +  2 * 32);   // col + 32
        v2f b3 = *(const v2f*)(wp +  3 * 32);   // col + 48
        acc0 = __builtin_amdgcn_wmma_f32_16x16x4_f32(false, a, false, b0, (short)0, acc0, false, false);
        acc1 = __builtin_amdgcn_wmma_f32_16x16x4_f32(false, a, false, b1, (short)0, acc1, false, false);
        acc2 = __builtin_amdgcn_wmma_f32_16x16x4_f32(false, a, false, b2, (short)0, acc2, false, false);
        acc3 = __builtin_amdgcn_wmma_f32_16x16x4_f32(false, a, false, b3, (short)0, acc3, false, false);
    }

    #pragma unroll
    for (int j = 0; j < 4; ++j) {
        v8f acc = (j == 0) ? acc0 : (j == 1) ? acc1 : (j == 2) ? acc2 : acc3;
        int zcol = nt4 * 64 + j * 16 + lm;   // 0..255
        float* Z = (zcol < HC) ? h : sk;
        int zc = zcol & (HC - 1);
        #pragma unroll
        for (int r = 0; r < 8; ++r) {
            int mm = m0 + r + (lh << 3);     // C/D layout: M = r + 8*half
            if (mm < Nn) Z[(size_t)mm * HC + zc] = acc[r];
        }
    }
}

// ---------- kernel 2: per-node attention dot products ----------
__global__ __launch_bounds__(256)
void att_dots(const float* __restrict__ h, const float* __restrict__ att_src,
              const float* __restrict__ att_dst,
              float* __restrict__ asrc, float* __restrict__ adst, int Nn)
{
    int node = blockIdx.x * 8 + (threadIdx.x >> 5);
    if (node >= Nn) return;
    int lane = threadIdx.x & 31;                 // lane covers channels lane*4..+3
    const float4 hv = *(const float4*)(h + (size_t)node * HC + lane * 4);
    const float4 as = *(const float4*)(att_src + lane * 4);
    const float4 ad = *(const float4*)(att_dst + lane * 4);
    float ps = hv.x*as.x + hv.y*as.y + hv.z*as.z + hv.w*as.w;
    float pd = hv.x*ad.x + hv.y*ad.y + hv.z*ad.z + hv.w*ad.w;
    ps += __shfl_xor(ps, 1, 32); ps += __shfl_xor(ps, 2, 32);   // 4 lanes = 1 head
    pd += __shfl_xor(pd, 1, 32); pd += __shfl_xor(pd, 2, 32);
    if ((lane & 3) == 0) {
        int hh = lane >> 2;
        asrc[(size_t)node * NHEAD + hh] = ps;
        adst[(size_t)node * NHEAD + hh] = pd;
    }
}

// ---------- kernel 3: segment max over destinations ----------
__global__ __launch_bounds__(256)
void edge_max(const int* __restrict__ ei, int E, int Nn,
              const float* __restrict__ asrc, const float* __restrict__ adst,
              unsigned* __restrict__ emax)
{
    int i = blockIdx.x * blockDim.x + threadIdx.x;
    if (i >= E + Nn) return;
    int s, d;
    if (i < E) { s = ei[i]; d = ei[E + i]; } else { s = d = i - E; }
    const float* As = asrc + (size_t)s * NHEAD;
    const float* Ad = adst + (size_t)d * NHEAD;
    unsigned* em = emax + (size_t)d * NHEAD;
    #pragma unroll
    for (int hh = 0; hh < NHEAD; ++hh)
        atomicMax(em + hh, encF(lrelu(As[hh] + Ad[hh])));
}

// ---------- kernel 4: segment exp-sum ----------
__global__ __launch_bounds__(256)
void edge_sum(const int* __restrict__ ei, int E, int Nn,
              const float* __restrict__ asrc, const float* __restrict__ adst,
              const unsigned* __restrict__ emax, float* __restrict__ denom)
{
    int i = blockIdx.x * blockDim.x + threadIdx.x;
    if (i >= E + Nn) return;
    int s, d;
    if (i < E) { s = ei[i]; d = ei[E + i]; } else { s = d = i - E; }
    const float* As = asrc + (size_t)s * NHEAD;
    const float* Ad = adst + (size_t)d * NHEAD;
    const unsigned* Em = emax + (size_t)d * NHEAD;
    float* Dn = denom + (size_t)d * NHEAD;
    #pragma unroll
    for (int hh = 0; hh < NHEAD; ++hh) {
        float e = lrelu(As[hh] + Ad[hh]);
        unsafeAtomicAdd(Dn + hh, __expf(e - decF(Em[hh])));
    }
}

// ---------- kernel 5: weighted scatter (dominant, BW-bound) ----------
// One wave per edge: lane owns 4 channels (head = lane>>2).
__global__ __launch_bounds__(256)
void edge_scatter(const int* __restrict__ ei, int E, int Nn,
                  const float* __restrict__ asrc, const float* __restrict__ adst,
                  const unsigned* __restrict__ emax, const float* __restrict__ denom,
                  const float* __restrict__ h, float* __restrict__ gat)
{
    int i = blockIdx.x * 8 + (threadIdx.x >> 5);
    if (i >= E + Nn) return;
    int lane = threadIdx.x & 31;
    int s, d;
    if (i < E) { s = ei[i]; d = ei[E + i]; } else { s = d = i - E; }
    int hh = lane >> 2;
    float e  = lrelu(asrc[(size_t)s * NHEAD + hh] + adst[(size_t)d * NHEAD + hh]);
    float mx = decF(emax[(size_t)d * NHEAD + hh]);
    float alpha = __expf(e - mx) / denom[(size_t)d * NHEAD + hh];
    const float4 hv = *(const float4*)(h + (size_t)s * HC + lane * 4);
    float* gd = gat + (size_t)d * HC + lane * 4;
    unsafeAtomicAdd(gd + 0, alpha * hv.x);
    unsafeAtomicAdd(gd + 1, alpha * hv.y);
    unsafeAtomicAdd(gd + 2, alpha * hv.z);
    unsafeAtomicAdd(gd + 3, alpha * hv.w);
}

// ---------- kernel 6: bias + skip + layernorm + elu ----------
__global__ __launch_bounds__(256)
void finalize_k(const float* __restrict__ gat, const float* __restrict__ sk,
                const float* __restrict__ bias, const float* __restrict__ gamma,
                const float* __restrict__ beta, float* __restrict__ out, int Nn)
{
    int node = blockIdx.x * 8 + (threadIdx.x >> 5);
    if (node >= Nn) return;
    int lane = threadIdx.x & 31;
    int c = lane * 4;
    const float4 g = *(const float4*)(gat + (size_t)node * HC + c);
    const float4 s = *(const float4*)(sk  + (size_t)node * HC + c);
    const float4 b = *(const float4*)(bias + c);
    float z0 = g.x + b.x + 0.1f * s.x;
    float z1 = g.y + b.y + 0.1f * s.y;
    float z2 = g.z + b.z + 0.1f * s.z;
    float z3 = g.w + b.w + 0.1f * s.w;

    float sum = z0 + z1 + z2 + z3;
    #pragma unroll
    for (int o = 1; o < 32; o <<= 1) sum += __shfl_xor(sum, o, 32);
    float mu = sum * (1.f / 128.f);

    float d0 = z0 - mu, d1 = z1 - mu, d2 = z2 - mu, d3 = z3 - mu;
    float vs = d0*d0 + d1*d1 + d2*d2 + d3*d3;
    #pragma unroll
    for (int o = 1; o < 32; o <<= 1) vs += __shfl_xor(vs, o, 32);
    float inv = rsqrtf(vs * (1.f / 128.f) + LN_EPS);

    const float4 gm = *(const float4*)(gamma + c);
    const float4 bt = *(const float4*)(beta  + c);
    float4 o4;
    o4.x = eluf(d0 * inv * gm.x + bt.x);
    o4.y = eluf(d1 * inv * gm.y + bt.y);
    o4.z = eluf(d2 * inv * gm.z + bt.z);
    o4.w = eluf(d3 * inv * gm.w + bt.w);
    *(float4*)(out + (size_t)node * HC + c) = o4;
}

// ---------- launch ----------
extern "C" void kernel_launch(void* const* d_in, const int* in_sizes, int n_in,
                              void* d_out, int out_size, void* d_ws, size_t ws_size,
                              hipStream_t stream)
{
    const float* x       = (const float*)d_in[0];
    const int*   ei      = (const int*)  d_in[1];
    const float* W_gat   = (const float*)d_in[2];
    const float* att_src = (const float*)d_in[3];
    const float* att_dst = (const float*)d_in[4];
    const float* bias    = (const float*)d_in[5];
    const float* W_skip  = (const float*)d_in[6];
    const float* gamma   = (const float*)d_in[7];
    const float* beta    = (const float*)d_in[8];

    int Nn = in_sizes[0] / F_IN;
    int E  = in_sizes[1] / 2;
    int EA = E + Nn;

    size_t off = 0;
    char* base = (char*)d_ws;
    auto alloc = [&](size_t bytes) -> void* {
        void* p = base + off;
        off += (bytes + 255) & ~(size_t)255;
        return p;
    };
    float*    h     = (float*)   alloc((size_t)Nn * HC * 4);
    float*    sk    = (float*)   alloc((size_t)Nn * HC * 4);
    float*    gat   = (float*)   alloc((size_t)Nn * HC * 4);
    float*    asrc  = (float*)   alloc((size_t)Nn * NHEAD * 4);
    float*    adst  = (float*)   alloc((size_t)Nn * NHEAD * 4);
    float*    denom = (float*)   alloc((size_t)Nn * NHEAD * 4);
    unsigned* emax  = (unsigned*)alloc((size_t)Nn * NHEAD * 4);
    float*    Wp    = (float*)   alloc((size_t)F_IN * 2 * HC * 4);

    int Mtiles = (Nn + 15) / 16;

    hipLaunchKernelGGL(init_ws, dim3((Nn * HC + 255) / 256), dim3(256), 0, stream,
                       gat, denom, emax, Nn);
    hipLaunchKernelGGL(pack_w, dim3((F_IN * 2 * HC + 255) / 256), dim3(256), 0, stream,
                       W_gat, W_skip, Wp);
    hipLaunchKernelGGL(gemm_dual, dim3((Mtiles * 4 + 7) / 8), dim3(256), 0, stream,
                       x, Wp, h, sk, Nn);
    hipLaunchKernelGGL(att_dots, dim3((Nn + 7) / 8), dim3(256), 0, stream,
                       h, att_src, att_dst, asrc, adst, Nn);
    hipLaunchKernelGGL(edge_max, dim3((EA + 255) / 256), dim3(256), 0, stream,
                       ei, E, Nn, asrc, adst, emax);
    hipLaunchKernelGGL(edge_sum, dim3((EA + 255) / 256), dim3(256), 0, stream,
                       ei, E, Nn, asrc, adst, emax, denom);
    hipLaunchKernelGGL(edge_scatter, dim3((EA + 7) / 8), dim3(256), 0, stream,
                       ei, E, Nn, asrc, adst, emax, denom, h, gat);
    hipLaunchKernelGGL(finalize_k, dim3((Nn + 7) / 8), dim3(256), 0, stream,
                       gat, sk, bias, gamma, beta, (float*)d_out, Nn);
}